// ProteinDNAGNN_59785944761269
// MI455X (gfx1250) — compile-verified
//
#include <hip/hip_runtime.h>
#include <hip/hip_bf16.h>

typedef __attribute__((ext_vector_type(2))) float v2f;
typedef __attribute__((ext_vector_type(8))) float v8f;

#define D_FEAT 128

// ---------------------------------------------------------------------------
// Utility: zero a float buffer
// ---------------------------------------------------------------------------
__global__ void zero_f32(float* __restrict__ p, int n) {
    int i = blockIdx.x * blockDim.x + threadIdx.x;
    if (i < n) p[i] = 0.0f;
}

// ---------------------------------------------------------------------------
// Degree: init to 1.0 (self-loop), count col occurrences, then rsqrt in place
// ---------------------------------------------------------------------------
__global__ void deg_init(float* __restrict__ deg, int n) {
    int i = blockIdx.x * blockDim.x + threadIdx.x;
    if (i < n) deg[i] = 1.0f;
}

__global__ void deg_count(const int* __restrict__ col, float* __restrict__ deg, int e) {
    int i = blockIdx.x * blockDim.x + threadIdx.x;
    if (i < e) atomicAdd(&deg[col[i]], 1.0f);
}

__global__ void deg_rsqrt(float* __restrict__ deg, int n) {
    int i = blockIdx.x * blockDim.x + threadIdx.x;
    if (i < n) deg[i] = rsqrtf(deg[i]);   // deg >= 1 guaranteed by self-loops
}

// ---------------------------------------------------------------------------
// GEMM: C[N x 128] = A[N x 128] * W[128 x 128], fp32 WMMA 16x16x4.
// Block = 256 threads = 8 waves. Each block: 16 rows of A, all 128 columns.
// Wave w computes the 16x16 tile at columns [16w, 16w+16). W staged in LDS.
//
// VGPR fragment layouts (CDNA5 ISA 7.12.2, 32-bit operands, wave32):
//   A 16x4 : lanes 0-15 -> M=lane, {v0,v1}={K0,K1}; lanes 16-31 -> {K2,K3}
//   B 4x16 : lanes 0-15 -> N=lane, {v0,v1}={K0,K1}; lanes 16-31 -> {K2,K3}
//   C 16x16: vr, lanes 0-15 -> (M=r,    N=lane)
//                lanes 16-31 -> (M=r+8,  N=lane-16)
// ---------------------------------------------------------------------------
__global__ void gemm_wmma_f32(const float* __restrict__ A,
                              const float* __restrict__ W,
                              float* __restrict__ C, int n) {
    __shared__ float sW[D_FEAT * D_FEAT];   // 64 KB

    const int tid = threadIdx.x;
    // Cooperative load of W into LDS (float4 per iteration)
    for (int i = tid; i < (D_FEAT * D_FEAT) / 4; i += 256) {
        ((float4*)sW)[i] = ((const float4*)W)[i];
    }
    __syncthreads();

    const int wave  = tid >> 5;
    const int lane  = tid & 31;
    const int lhalf = lane >> 4;   // 0: K pair {0,1}; 1: K pair {2,3}
    const int lmod  = lane & 15;

    const int row0 = blockIdx.x * 16;
    const int col0 = wave * 16;

    int arow = row0 + lmod;
    if (arow >= n) arow = n - 1;   // clamp (no divergence; EXEC stays all-1 for WMMA)

    v8f acc = {};
    for (int k0 = 0; k0 < D_FEAT; k0 += 4) {
        const float* ap = A + (size_t)arow * D_FEAT + (k0 + lhalf * 2);
        v2f a;
        a.x = ap[0];
        a.y = ap[1];
        const float* bp = sW + (size_t)(k0 + lhalf * 2) * D_FEAT + (col0 + lmod);
        v2f b;
        b.x = bp[0];
        b.y = bp[D_FEAT];
        acc = __builtin_amdgcn_wmma_f32_16x16x4_f32(
            /*neg_a=*/false, a, /*neg_b=*/false, b,
            /*c_mod=*/(short)0, acc, /*reuse_a=*/false, /*reuse_b=*/false);
    }

    const int orow = row0 + lhalf * 8;
    float* cp = C + (size_t)orow * D_FEAT + (col0 + lmod);
#pragma unroll
    for (int r = 0; r < 8; ++r) {
        if (orow + r < n) cp[(size_t)r * D_FEAT] = acc[r];
    }
}

// ---------------------------------------------------------------------------
// Edge scatter: one wave per edge; lane handles a 16B chunk of the 512B row.
// acc[col] += h[row] * (dinv[row]*dinv[col])
// h (25.6 MB) is fully L2-resident -> gather and f32 atomics run at L2 rates.
// ---------------------------------------------------------------------------
__global__ void scatter_edges(const int* __restrict__ rows,
                              const int* __restrict__ cols,
                              const float* __restrict__ dinv,
                              const float* __restrict__ h,
                              float* __restrict__ acc, int e) {
    int idx  = blockIdx.x * blockDim.x + threadIdx.x;
    int edge = idx >> 5;
    if (edge >= e) return;
    int lane = idx & 31;
    int r = rows[edge];
    int c = cols[edge];
    float nrm = dinv[r] * dinv[c];
    const float4 hv = *(const float4*)(h + (size_t)r * D_FEAT + lane * 4);
    float* outp = acc + (size_t)c * D_FEAT + lane * 4;
    atomicAdd(outp + 0, hv.x * nrm);
    atomicAdd(outp + 1, hv.y * nrm);
    atomicAdd(outp + 2, hv.z * nrm);
    atomicAdd(outp + 3, hv.w * nrm);
}

// ---------------------------------------------------------------------------
// Epilogue: out = relu(acc + selfloop(t) + bias). Self-loop term = t * dinv^2.
// Safe when out aliases t (pure elementwise, same index read then written).
// ---------------------------------------------------------------------------
__global__ void finish_layer(const float* __restrict__ acc,
                             const float* __restrict__ t,
                             const float* __restrict__ dinv,
                             const float* __restrict__ bias,
                             float* __restrict__ out, int n) {
    int idx = blockIdx.x * blockDim.x + threadIdx.x;
    if (idx >= n * D_FEAT) return;
    int node = idx >> 7;
    int f    = idx & (D_FEAT - 1);
    float d  = dinv[node];
    float v  = acc[idx] + t[idx] * d * d + bias[f];
    out[idx] = fmaxf(v, 0.0f);
}

// ---------------------------------------------------------------------------
// Global mean-pool part 1: atomic segment-sum into sums[G][128], counts[G].
// One wave per node; lane handles 4 features.
// ---------------------------------------------------------------------------
__global__ void pool_sum(const float* __restrict__ h,
                         const int* __restrict__ batch,
                         float* __restrict__ sums,
                         float* __restrict__ counts, int n) {
    int idx  = blockIdx.x * blockDim.x + threadIdx.x;
    int node = idx >> 5;
    if (node >= n) return;
    int lane = idx & 31;
    int g = batch[node];
    const float4 hv = *(const float4*)(h + (size_t)node * D_FEAT + lane * 4);
    float* sp = sums + (size_t)g * D_FEAT + lane * 4;
    atomicAdd(sp + 0, hv.x);
    atomicAdd(sp + 1, hv.y);
    atomicAdd(sp + 2, hv.z);
    atomicAdd(sp + 3, hv.w);
    if (lane == 0) atomicAdd(&counts[g], 1.0f);
}

// ---------------------------------------------------------------------------
// Final FC: out[g] = dot(sums[g]/count[g], fc_w) + fc_b. One wave per graph.
// ---------------------------------------------------------------------------
__global__ void fc_out(const float* __restrict__ sums,
                       const float* __restrict__ counts,
                       const float* __restrict__ fcw,
                       const float* __restrict__ fcb,
                       float* __restrict__ out, int g_count) {
    int idx  = blockIdx.x * blockDim.x + threadIdx.x;
    int g    = idx >> 5;
    int lane = idx & 31;
    if (g >= g_count) return;
    float s = 0.0f;
    for (int f = lane; f < D_FEAT; f += 32)
        s += sums[(size_t)g * D_FEAT + f] * fcw[f];
#pragma unroll
    for (int o = 16; o > 0; o >>= 1)
        s += __shfl_down(s, o, 32);
    if (lane == 0) {
        float cnt = fmaxf(counts[g], 1.0f);
        out[g] = s / cnt + fcb[0];
    }
}

// ---------------------------------------------------------------------------
// Launch
// ---------------------------------------------------------------------------
extern "C" void kernel_launch(void* const* d_in, const int* in_sizes, int n_in,
                              void* d_out, int out_size, void* d_ws, size_t ws_size,
                              hipStream_t stream) {
    (void)n_in; (void)ws_size;

    const float* x     = (const float*)d_in[0];
    const int*   ei    = (const int*)d_in[1];
    const int*   batch = (const int*)d_in[2];
    const float* Wl[3] = {(const float*)d_in[3], (const float*)d_in[5], (const float*)d_in[7]};
    const float* Bl[3] = {(const float*)d_in[4], (const float*)d_in[6], (const float*)d_in[8]};
    const float* fcw   = (const float*)d_in[9];
    const float* fcb   = (const float*)d_in[10];
    float* out = (float*)d_out;

    const int N = in_sizes[0] / D_FEAT;   // 50000
    const int E = in_sizes[1] / 2;        // 800000
    const int G = out_size;               // 64

    // Workspace carve-up (~77 MB total)
    auto aup = [](size_t v) { return (v + 255) & ~(size_t)255; };
    char* p = (char*)d_ws;
    float* dinv   = (float*)p; p += aup((size_t)N * 4);
    float* bufA   = (float*)p; p += aup((size_t)N * D_FEAT * 4);
    float* bufB   = (float*)p; p += aup((size_t)N * D_FEAT * 4);
    float* bufC   = (float*)p; p += aup((size_t)N * D_FEAT * 4);
    float* sums   = (float*)p; p += aup((size_t)G * D_FEAT * 4);
    float* counts = (float*)p;

    const int* erow = ei;       // edge_index[0] = source nodes
    const int* ecol = ei + E;   // edge_index[1] = destination nodes

    const int TB = 256;
    const int nElem = N * D_FEAT;

    // --- normalization coefficients ---
    deg_init <<<(N + TB - 1) / TB, TB, 0, stream>>>(dinv, N);
    deg_count<<<(E + TB - 1) / TB, TB, 0, stream>>>(ecol, dinv, E);
    deg_rsqrt<<<(N + TB - 1) / TB, TB, 0, stream>>>(dinv, N);

    // --- layer schedule: {in, t(gemm out), acc, out} ---
    const float* lin[3] = {x,    bufC, bufA};
    float*       lt [3] = {bufA, bufA, bufB};
    float*       lac[3] = {bufB, bufB, bufC};
    float*       lo [3] = {bufC, bufA, bufB};

    const int gemmBlocks    = (N + 15) / 16;
    const int scatterBlocks = (int)(((size_t)E * 32 + TB - 1) / TB);

    for (int l = 0; l < 3; ++l) {
        gemm_wmma_f32<<<gemmBlocks, TB, 0, stream>>>(lin[l], Wl[l], lt[l], N);
        zero_f32     <<<(nElem + TB - 1) / TB, TB, 0, stream>>>(lac[l], nElem);
        scatter_edges<<<scatterBlocks, TB, 0, stream>>>(erow, ecol, dinv, lt[l], lac[l], E);
        finish_layer <<<(nElem + TB - 1) / TB, TB, 0, stream>>>(lac[l], lt[l], dinv, Bl[l], lo[l], N);
    }

    // --- global mean pool + FC ---
    const int poolN = G * D_FEAT + G;
    zero_f32<<<(poolN + TB - 1) / TB, TB, 0, stream>>>(sums, G * D_FEAT);
    zero_f32<<<1, TB, 0, stream>>>(counts, G);
    pool_sum<<<(int)(((size_t)N * 32 + TB - 1) / TB), TB, 0, stream>>>(lo[2], batch, sums, counts, N);
    fc_out  <<<(G * 32 + TB - 1) / TB, TB, 0, stream>>>(sums, counts, fcw, fcb, out, G);
}